// YOLOv1loss_16303695855705
// MI455X (gfx1250) — compile-verified
//
#include <hip/hip_runtime.h>

// ---------------------------------------------------------------------------
// YOLOv1 loss, MI455X (gfx1250, wave32).
// Memory-bound streaming reduction: 386 MB @ 23.3 TB/s => ~16.6 us floor.
// Wave-level reduction via V_WMMA_F32_16X16X4_F32 (A = 64 wave partials,
// B = ones). Raw v_sqrt_f32 / v_rcp_f32 to strip IEEE fixup VALU chains and
// keep the loop safely bandwidth-bound.
// ---------------------------------------------------------------------------

typedef float v2f __attribute__((ext_vector_type(2)));
typedef float v8f __attribute__((ext_vector_type(8)));

#define BLOCK 256  // 8 wave32s per workgroup

// Sum the 64 f32 accumulators of a wave (2 per lane) with one WMMA.
// A (16x4 f32): lane L<16 holds A[L][0],A[L][1]; lane 16+L holds A[L][2],A[L][3].
// With B = ones, D[m][n] = sum_k A[m][k]. D layout: lanes 0-15 VGPR r = row r,
// lanes 16-31 VGPR r = row 8+r. Sum 8 components + shfl_xor(16) => wave total.
__device__ __forceinline__ float wave_sum_wmma(float a0, float a1) {
  v2f A;  A.x = a0;  A.y = a1;
  v2f Bm; Bm.x = 1.0f; Bm.y = 1.0f;
  v8f C = {};
  v8f D = __builtin_amdgcn_wmma_f32_16x16x4_f32(
      /*neg_a=*/false, A, /*neg_b=*/false, Bm,
      /*c_mod=*/(short)0, C, /*reuse_a=*/false, /*reuse_b=*/false);
  float s = D[0] + D[1] + D[2] + D[3] + D[4] + D[5] + D[6] + D[7];
  s += __shfl_xor(s, 16, 32);
  return s;  // full wave sum, broadcast to every lane
}

__device__ __forceinline__ float block_sum(float a0, float a1) {
  __shared__ float lds[BLOCK / 32];
  float w = wave_sum_wmma(a0, a1);
  int lane = threadIdx.x & 31;
  int wid  = threadIdx.x >> 5;
  if (lane == 0) lds[wid] = w;
  __syncthreads();
  float r = 0.0f;
  if (threadIdx.x == 0) {
#pragma unroll
    for (int i = 0; i < BLOCK / 32; ++i) r += lds[i];
  }
  return r;  // valid in thread 0 only
}

__device__ __forceinline__ float iou_box(float px1, float py1, float px2, float py2,
                                         float lx1, float ly1, float lx2, float ly2) {
  float ix1 = fmaxf(px1, lx1), iy1 = fmaxf(py1, ly1);
  float ix2 = fminf(px2, lx2), iy2 = fminf(py2, ly2);
  float iw = fmaxf(ix2 - ix1 + 1.0f, 0.0f);
  float ih = fmaxf(iy2 - iy1 + 1.0f, 0.0f);
  float inter = iw * ih;
  float a1 = (px2 - px1 + 1.0f) * (py2 - py1 + 1.0f);
  float a2 = (lx2 - lx1 + 1.0f) * (ly2 - ly1 + 1.0f);
  // bare v_rcp_f32 + v_mul: data domain keeps denom well away from 0/denorm
  return inter * __builtin_amdgcn_rcpf(a1 + a2 - inter);
}

__global__ __launch_bounds__(BLOCK) void yolo_partial(
    const float* __restrict__ pre, const float* __restrict__ lab,
    float* __restrict__ partials, int ncells) {
  const int tid    = blockIdx.x * BLOCK + threadIdx.x;
  const int stride = gridDim.x * BLOCK;
  float acc0 = 0.0f, acc1 = 0.0f;
  int flip = 0;

  for (int i = tid; i < ncells; i += stride) {
    const int b = i / 49;
    const int r = i - b * 49;
    const float* pb = pre + b * 1470 + r;  // [B,30,7,7]: chan stride 49
    const float* lb = lab + b * 1470 + r;

    float p[10], l[10];
#pragma unroll
    for (int c = 0; c < 10; ++c) {
      p[c] = __builtin_nontemporal_load(pb + c * 49);
      l[c] = __builtin_nontemporal_load(lb + c * 49);
    }
    // class loss: every cell, channels 10..29
    float cls = 0.0f;
#pragma unroll
    for (int c = 10; c < 30; ++c) {
      float d = __builtin_nontemporal_load(pb + c * 49) -
                __builtin_nontemporal_load(lb + c * 49);
      cls = fmaf(d, d, cls);
    }

    float iou1 = iou_box(p[0], p[1], p[2], p[3], l[0], l[1], l[2], l[3]);
    float iou2 = iou_box(p[5], p[6], p[7], p[8], l[5], l[6], l[7], l[8]);
    bool resp1 = iou1 >= iou2;

    float dx1 = l[0] - p[0], dy1 = l[1] - p[1];
    float addr1 = 5.0f * (dx1 * dx1 + dy1 * dy1);
    float dx2 = l[5] - p[5], dy2 = l[6] - p[6];
    float addr2 = 5.0f * (dx2 * dx2 + dy2 * dy2);

    // bare v_sqrt_f32: inputs in [0.05, 1), no denorm/negative fixup needed
    float sa = __builtin_amdgcn_sqrtf(p[2]) - __builtin_amdgcn_sqrtf(l[2]);
    float sb = __builtin_amdgcn_sqrtf(p[3]) - __builtin_amdgcn_sqrtf(l[3]);
    float size1 = 5.0f * (sa * sa + sb * sb);
    float sc = __builtin_amdgcn_sqrtf(p[7]) - __builtin_amdgcn_sqrtf(l[7]);
    float sd = __builtin_amdgcn_sqrtf(p[8]) - __builtin_amdgcn_sqrtf(l[8]);
    float size2 = 5.0f * (sc * sc + sd * sd);

    float ciou = resp1 ? iou1 : iou2;
    float cd   = p[9] - ciou;
    float conf = 0.5f * cd * cd;

    float objt  = (resp1 ? (addr1 + size1) : (addr2 + size2)) + conf;
    float nob   = p[4] + p[9];
    float noobj = 0.5f * nob * nob;

    float term = ((l[4] == 1.0f) ? objt : noobj) + cls;
    if (flip) acc1 += term; else acc0 += term;
    flip ^= 1;
  }

  float bs = block_sum(acc0, acc1);
  if (threadIdx.x == 0) partials[blockIdx.x] = bs;
}

__global__ __launch_bounds__(BLOCK) void yolo_final(
    const float* __restrict__ partials, int n, float* __restrict__ out,
    float scale) {
  float acc0 = 0.0f, acc1 = 0.0f;
  int flip = 0;
  for (int i = threadIdx.x; i < n; i += BLOCK) {
    float v = partials[i];
    if (flip) acc1 += v; else acc0 += v;
    flip ^= 1;
  }
  float t = block_sum(acc0, acc1);
  if (threadIdx.x == 0) out[0] = t * scale;
}

extern "C" void kernel_launch(void* const* d_in, const int* in_sizes, int n_in,
                              void* d_out, int out_size, void* d_ws, size_t ws_size,
                              hipStream_t stream) {
  const float* pre = (const float*)d_in[0];
  const float* lab = (const float*)d_in[1];
  float* out      = (float*)d_out;
  float* partials = (float*)d_ws;

  const int batch  = in_sizes[0] / (30 * 49);  // 32768
  const int ncells = batch * 49;

  int nblocks = (ncells + BLOCK - 1) / BLOCK;  // 6272 for B=32768
  int maxblocks = (int)(ws_size / sizeof(float));
  if (maxblocks > 0 && nblocks > maxblocks) nblocks = maxblocks;
  if (nblocks < 1) nblocks = 1;

  yolo_partial<<<nblocks, BLOCK, 0, stream>>>(pre, lab, partials, ncells);
  yolo_final<<<1, BLOCK, 0, stream>>>(partials, nblocks, out,
                                      1.0f / (float)batch);
}